// POIEncoder_66133906424032
// MI455X (gfx1250) — compile-verified
//
#include <hip/hip_runtime.h>
#include <hip/hip_bf16.h>

typedef __attribute__((ext_vector_type(2))) float v2f;
typedef __attribute__((ext_vector_type(8))) float v8f;

#define HID 128
#define IN_C 128

// ---------------- degree / norm ----------------
__global__ void deg_init_kernel(float* __restrict__ deg, int n) {
    int i = blockIdx.x * blockDim.x + threadIdx.x;
    if (i < n) deg[i] = 1.0f;  // self-loop weight
}

__global__ void deg_acc_kernel(const long long* __restrict__ col,
                               const float* __restrict__ ew,
                               float* __restrict__ deg, int E) {
    int e = blockIdx.x * blockDim.x + threadIdx.x;
    if (e < E) {
        unsafeAtomicAdd(&deg[(size_t)col[e]], ew[e]);
    }
}

__global__ void deg_rsqrt_kernel(float* __restrict__ deg, int n) {
    int i = blockIdx.x * blockDim.x + threadIdx.x;
    if (i < n) {
        float d = deg[i];
        deg[i] = (d > 0.0f) ? rsqrtf(d) : 0.0f;  // now holds dinv
    }
}

// ---------------- h = x @ W^T via fp32 WMMA (16x16x4) ----------------
// One wave computes a 16(M) x 16(N) tile of h, looping K=128 in steps of 4.
// A-frag (16x4 f32): lane l (0-15) holds K=0,1; lane l+16 holds K=2,3.
// B-frag (4x16 f32): B[k][n] = W[n][k]; same half-wave K split.
// C/D (16x16 f32, 8 VGPRs): vgpr r, lanes 0-15 -> M=r, lanes 16-31 -> M=r+8.
__global__ __launch_bounds__(256)
void gemm_wmma_kernel(const float* __restrict__ x,
                      const float* __restrict__ W,
                      float* __restrict__ h, int n_rows) {
    const int lane  = threadIdx.x & 31;
    const int wave  = threadIdx.x >> 5;   // 0..7 -> N-tile (HID/16 = 8)
    const int half  = lane >> 4;          // 0 or 1
    const int l     = lane & 15;
    const int tileM = blockIdx.x;

    int rowA = tileM * 16 + l;
    if (rowA >= n_rows) rowA = n_rows - 1;     // clamp reads (keep EXEC all-1 for WMMA)
    const int colB = wave * 16 + l;            // output channel / W row

    const float* __restrict__ xrow = x + (size_t)rowA * IN_C;
    const float* __restrict__ wrow = W + (size_t)colB * IN_C;

    v8f c = {};
#pragma unroll 8
    for (int k0 = 0; k0 < IN_C; k0 += 4) {
        v2f a, b;
        const int ka = k0 + 2 * half;
        a.x = xrow[ka + 0];
        a.y = xrow[ka + 1];
        b.x = wrow[ka + 0];
        b.y = wrow[ka + 1];
        // D = A * B + C   (emits v_wmma_f32_16x16x4_f32)
        c = __builtin_amdgcn_wmma_f32_16x16x4_f32(
            /*neg_a=*/false, a, /*neg_b=*/false, b,
            /*c_mod=*/(short)0, c, /*reuse_a=*/false, /*reuse_b=*/false);
    }

    // Epilogue: single tile-level bounds check; unguarded straight-line
    // coalesced stores on the (always-taken for N%16==0) fast path.
    const int mBase = tileM * 16 + 8 * half;
    float* __restrict__ hbase = h + (size_t)mBase * HID + wave * 16 + l;
    if (tileM * 16 + 16 <= n_rows) {
#pragma unroll
        for (int r = 0; r < 8; ++r) {
            hbase[(size_t)r * HID] = c[r];
        }
    } else {
#pragma unroll
        for (int r = 0; r < 8; ++r) {
            if (mBase + r < n_rows) hbase[(size_t)r * HID] = c[r];
        }
    }
}

// ---------------- out = bias + self-loop term ----------------
__global__ void out_init_kernel(const float* __restrict__ h,
                                const float* __restrict__ dinv,
                                const float* __restrict__ bias,
                                float* __restrict__ out, int total) {
    int idx = blockIdx.x * blockDim.x + threadIdx.x;
    if (idx < total) {
        const int i = idx >> 7;       // node
        const int cch = idx & 127;    // channel
        const float di = dinv[i];
        out[idx] = bias[cch] + di * di * h[idx];
    }
}

// ---------------- edge scatter: one wave per edge ----------------
// lane handles 4 contiguous channels: coalesced 512B float4 gather from h,
// 4 native f32 atomic adds into out (out + h both L2-resident: 51MB each).
__global__ __launch_bounds__(256)
void scatter_kernel(const long long* __restrict__ row,
                    const long long* __restrict__ col,
                    const float* __restrict__ ew,
                    const float* __restrict__ dinv,
                    const float* __restrict__ h,
                    float* __restrict__ out, int E) {
    const int wavesPerBlock = blockDim.x >> 5;
    const long long e = (long long)blockIdx.x * wavesPerBlock + (threadIdx.x >> 5);
    if (e >= E) return;
    const int lane = threadIdx.x & 31;

    const long long r = row[e];
    const long long t = col[e];
    const float nrm = dinv[r] * ew[e] * dinv[t];

    const float4 hv = *(const float4*)(h + (size_t)r * HID + lane * 4);
    float* ob = out + (size_t)t * HID + lane * 4;
    unsafeAtomicAdd(ob + 0, nrm * hv.x);
    unsafeAtomicAdd(ob + 1, nrm * hv.y);
    unsafeAtomicAdd(ob + 2, nrm * hv.z);
    unsafeAtomicAdd(ob + 3, nrm * hv.w);
}

// ---------------- PReLU ----------------
__global__ void prelu_kernel(float* __restrict__ out,
                             const float* __restrict__ alpha, int total) {
    int idx = blockIdx.x * blockDim.x + threadIdx.x;
    if (idx < total) {
        const float v = out[idx];
        const int cch = idx & 127;
        out[idx] = (v > 0.0f) ? v : alpha[cch] * v;
    }
}

extern "C" void kernel_launch(void* const* d_in, const int* in_sizes, int n_in,
                              void* d_out, int out_size, void* d_ws, size_t ws_size,
                              hipStream_t stream) {
    const float*      x     = (const float*)d_in[0];
    const long long*  ei    = (const long long*)d_in[1];   // [2, E] int64
    const float*      ew    = (const float*)d_in[2];
    const float*      W     = (const float*)d_in[3];
    const float*      bias  = (const float*)d_in[4];
    const float*      alpha = (const float*)d_in[5];
    float*            out   = (float*)d_out;

    const int N = in_sizes[0] / IN_C;
    const int E = in_sizes[2];
    const long long* row = ei;       // source
    const long long* col = ei + E;   // target

    // workspace: h [N*HID floats] then dinv [N floats]
    float* h    = (float*)d_ws;
    float* dinv = h + (size_t)N * HID;

    const int total = N * HID;

    // 1) deg = 1.0 (self-loops)
    deg_init_kernel<<<(N + 255) / 256, 256, 0, stream>>>(dinv, N);
    // 2) deg[col] += ew
    deg_acc_kernel<<<(E + 255) / 256, 256, 0, stream>>>(col, ew, dinv, E);
    // 3) dinv = rsqrt(deg)
    deg_rsqrt_kernel<<<(N + 255) / 256, 256, 0, stream>>>(dinv, N);
    // 4) h = x @ W^T (fp32 WMMA)
    gemm_wmma_kernel<<<(N + 15) / 16, 256, 0, stream>>>(x, W, h, N);
    // 5) out = bias + dinv^2 * h   (self-loop message folded in)
    out_init_kernel<<<(total + 255) / 256, 256, 0, stream>>>(h, dinv, bias, out, total);
    // 6) out[col] += dinv[row]*ew*dinv[col] * h[row]
    scatter_kernel<<<(E + 7) / 8, 256, 0, stream>>>(row, col, ew, dinv, h, out, E);
    // 7) PReLU
    prelu_kernel<<<(total + 255) / 256, 256, 0, stream>>>(out, alpha, total);
}